// Model_70660801954261
// MI455X (gfx1250) — compile-verified
//
#include <hip/hip_runtime.h>
#include <hip/hip_bf16.h>

// ---- Problem constants (from reference) ----
#define BATCH   64
#define CHAN    32
#define BC      2048          // BATCH*CHAN
#define SEQ     512
#define MAXP    32
#define RR      16            // SEQ/MAXP
#define TT      4             // MAXP/MINP
#define DM      256           // D_MODEL
#define NPAIR   (BC*RR)       // 32768 (bc,r) pairs
#define OUT_FLOATS ((size_t)BC*RR*TT*DM)   // 33,554,432 f32 (~128MB)

typedef __attribute__((ext_vector_type(2))) float v2f;
typedef __attribute__((ext_vector_type(8))) float v8f;

// -ln(10000)/256
#define NEG_LOG1E4_OVER_D (-0.035977892f)
// exp(16 * NEG_LOG1E4_OVER_D) = 10000^(-16/256)
#define FAC_STEP 0.56234133f
#define HALF_PI  1.57079633f

// ---------------------------------------------------------------------------
// Kernel 1: per-(bc,r) classifier.  h = relu(x32 @ w1 + b1); logits = h@w2+b2;
// argmax -> expert id.  Written to the int32 tail of d_out in TRANSPOSED
// order (all_cls_pred[r*BC + bc]), exactly as the reference returns it.
// ---------------------------------------------------------------------------
__global__ __launch_bounds__(256) void cls_kernel(
    const float* __restrict__ x,  const float* __restrict__ w1,
    const float* __restrict__ b1, const float* __restrict__ w2,
    const float* __restrict__ b2, int* __restrict__ pred_out)
{
    __shared__ float sW1[32 * 64];
    __shared__ float sB1[64];
    __shared__ float sW2[64 * 3];
    __shared__ float sB2[3];
    const int tid = threadIdx.x;
    for (int i = tid; i < 32 * 64; i += 256) sW1[i] = w1[i];
    for (int i = tid; i < 64 * 3;  i += 256) sW2[i] = w2[i];
    if (tid < 64) sB1[tid] = b1[tid];
    if (tid < 3)  sB2[tid] = b2[tid];
    __syncthreads();

    const int g = blockIdx.x * 256 + tid;          // pair id = bc*16 + r
    if (g >= NPAIR) return;
    const int bc = g >> 4, r = g & 15;
    const float* xrow = x + (size_t)bc * SEQ + r * MAXP;

    float xv[32];
#pragma unroll
    for (int k = 0; k < 32; ++k) xv[k] = xrow[k];

    float l0 = sB2[0], l1 = sB2[1], l2 = sB2[2];
#pragma unroll 4
    for (int j = 0; j < 64; ++j) {
        float acc = sB1[j];
#pragma unroll
        for (int k = 0; k < 32; ++k) acc = fmaf(xv[k], sW1[k * 64 + j], acc);
        const float h = fmaxf(acc, 0.0f);
        l0 = fmaf(h, sW2[j * 3 + 0], l0);
        l1 = fmaf(h, sW2[j * 3 + 1], l1);
        l2 = fmaf(h, sW2[j * 3 + 2], l2);
    }
    // first-max semantics of jnp.argmax
    int best = 0;
    float bv = l0;
    if (l1 > bv) { best = 1; bv = l1; }
    if (l2 > bv) { best = 2; }
    pred_out[r * BC + bc] = best;                  // transposed layout
}

// ---------------------------------------------------------------------------
// Kernel 2: mixture-of-patch-embeddings via V_WMMA_F32_16X16X4_F32.
// Each wave does 8 (bc,r) pairs, one at a time.  O(4x256) = A(4x32, K zero-padded) @ W_e(32x256).
//   A-fragment: lane m(0-15) VGPR{0,1} = A[m, 4kk+{0,1}], lane 16+m = K+{2,3}
//   B-fragment: lane l(0-15) = W[4kk+{0,1}, 16nn+l], lane 16+l = K+{2,3}
//   D: acc[t], lanes 0-15 hold rows t=0..3 (col = lane).
// Embedding tables staged compactly in 56KB LDS (reads past an expert's p
// rows hit the next table — finite garbage x 0 = 0).
// PE uses a single branch-free v_sin per element: cos(x)=sin(x+pi/2) with a
// per-lane phase, and the frequency factor updated by one multiply per
// N-tile (fac *= 10000^(-16/256)) instead of exp() each iteration.
// ---------------------------------------------------------------------------
__global__ __launch_bounds__(256) void embed_kernel(
    const float* __restrict__ x,
    const float* __restrict__ we0, const float* __restrict__ we1,
    const float* __restrict__ we2,
    const int* __restrict__ pred, float* __restrict__ out)
{
    __shared__ float sEmb[56 * 256];               // 8+16+32 rows of 256, 56KB
    const int tid = threadIdx.x;
    for (int i = tid; i < 56 * 256; i += 256) {
        const int d = i & 255;
        const int kr = i >> 8;                     // 0..55
        float v;
        if (kr < 8)       v = we0[i];
        else if (kr < 24) v = we1[(kr - 8) * 256 + d];
        else              v = we2[(kr - 24) * 256 + d];
        sEmb[i] = v;
    }
    __syncthreads();

    const int lane   = tid & 31;
    const int half   = lane >> 4;                  // 0: K={0,1}, 1: K={2,3}
    const int lane16 = lane & 15;
    const int gwave  = blockIdx.x * 8 + (tid >> 5);   // 0..4095

    // Per-lane PE constants (column d = 16*nn + lane16):
    //   pe(pos,d) = sin(pos * exp(c*(d&~1)) + phase),  phase = (d&1)*pi/2
    const float phase   = (lane16 & 1) ? HALF_PI : 0.0f;
    const float facLane = __expf(NEG_LOG1E4_OVER_D * (float)(lane16 & ~1));

    for (int it = 0; it < 8; ++it) {
        const int g  = gwave + it * 4096;          // pair id
        const int bc = g >> 4, r = g & 15;
        const int e  = pred[r * BC + bc];
        const int p  = 8 << e;                     // 8 / 16 / 32
        const float* xrow = x + (size_t)bc * SEQ + r * MAXP;
        const float* embp = sEmb + ((e == 0) ? 0 : (e == 1) ? 8 * 256 : 24 * 256);

        // Build A fragments for all 8 K-steps (hoisted out of the N loop).
        v2f afrag[8];
        const int m = lane16;                      // output row (t), valid < 4
#pragma unroll
        for (int kk = 0; kk < 8; ++kk) {
            float a0 = 0.0f, a1 = 0.0f;
            if (m < 4) {
                const int t   = m;
                const int off = (p == 8) ? t * 8 : (p == 16) ? ((t == 3) ? 16 : 0) : 0;
                const int k0  = kk * 4 + half * 2;
                if (k0 < p)     a0 = xrow[off + k0];
                if (k0 + 1 < p) a1 = xrow[off + k0 + 1];
            }
            afrag[kk].x = a0; afrag[kk].y = a1;
        }

        float* outbase = out + ((size_t)bc * 64 + (size_t)r * 4) * DM;
        float fac = facLane;                       // frequency for column d

        for (int nn = 0; nn < 16; ++nn) {
            const int d = nn * 16 + lane16;        // output column
            v8f acc = {};
#pragma unroll
            for (int kk = 0; kk < 8; ++kk) {
                const int kb = kk * 4 + half * 2;
                v2f b;
                b.x = embp[kb * 256 + d];
                b.y = embp[(kb + 1) * 256 + d];
                acc = __builtin_amdgcn_wmma_f32_16x16x4_f32(
                    /*neg_a=*/false, afrag[kk], /*neg_b=*/false, b,
                    /*c_mod=*/(short)0, acc, /*reuse_a=*/false, /*reuse_b=*/false);
            }
            if (half == 0) {                       // rows t=0..3 live on lanes 0-15
#pragma unroll
                for (int t = 0; t < 4; ++t) {
                    const float pe = __sinf(fmaf((float)(r * 4 + t), fac, phase));
                    outbase[(size_t)t * DM + d] = acc[t] + pe;
                }
            }
            fac *= FAC_STEP;                       // advance to next N-tile
        }
    }
}

extern "C" void kernel_launch(void* const* d_in, const int* in_sizes, int n_in,
                              void* d_out, int out_size, void* d_ws, size_t ws_size,
                              hipStream_t stream) {
    const float* x   = (const float*)d_in[0];
    const float* w1  = (const float*)d_in[1];
    const float* b1  = (const float*)d_in[2];
    const float* w2  = (const float*)d_in[3];
    const float* b2  = (const float*)d_in[4];
    const float* we0 = (const float*)d_in[5];
    const float* we1 = (const float*)d_in[6];
    const float* we2 = (const float*)d_in[7];

    float* out  = (float*)d_out;
    int*   pred = (int*)(out + OUT_FLOATS);        // int32 tail of tuple output

    cls_kernel<<<NPAIR / 256, 256, 0, stream>>>(x, w1, b1, w2, b2, pred);
    embed_kernel<<<512, 256, 0, stream>>>(x, we0, we1, we2, pred, out);
}